// GCNNet_25340307046788
// MI455X (gfx1250) — compile-verified
//
#include <hip/hip_runtime.h>

#define N_NODES  100000
#define N_EDGES  1600000
#define N_GRAPHS 512
#define IN_CH    3
#define H1C      64
#define H2C      128
#define OUT_CH   2

typedef __attribute__((ext_vector_type(2))) float v2f;
typedef __attribute__((ext_vector_type(8))) float v8f;

// ---------------- small utility kernels ----------------

__global__ void fill_f32(float* __restrict__ p, float v, int n) {
    int i = blockIdx.x * blockDim.x + threadIdx.x;
    if (i < n) p[i] = v;
}

__global__ void deg_accum(const int* __restrict__ dst, float* __restrict__ deg) {
    int e = blockIdx.x * blockDim.x + threadIdx.x;
    if (e < N_EDGES) atomicAdd(&deg[dst[e]], 1.0f);
}

__global__ void deg_to_dinv(float* __restrict__ d) {
    int n = blockIdx.x * blockDim.x + threadIdx.x;
    if (n < N_NODES) d[n] = rsqrtf(d[n]);
}

// xw1 = x @ W1   (N x 3) @ (3 x 64); thread per (node, channel)
__global__ void xw1_kernel(const float* __restrict__ x, const float* __restrict__ W1,
                           float* __restrict__ xw1) {
    int i = blockIdx.x * blockDim.x + threadIdx.x;
    if (i >= N_NODES * H1C) return;
    int n = i >> 6, c = i & 63;
    float acc = 0.f;
#pragma unroll
    for (int k = 0; k < IN_CH; ++k) acc += x[n * IN_CH + k] * W1[k * H1C + c];
    xw1[i] = acc;
}

// h[n,c] = dinv[n]^2 * xw[n,c]   (self-loop term, init of aggregation buffer)
template <int NCH>
__global__ void self_init(const float* __restrict__ xw, const float* __restrict__ dinv,
                          float* __restrict__ h) {
    int i = blockIdx.x * blockDim.x + threadIdx.x;
    if (i >= N_NODES * NCH) return;
    int n = i / NCH;
    float di = dinv[n];
    h[i] = di * di * xw[i];
}

// per-edge scatter: h[dst,c] += dinv[src]*dinv[dst] * xw[src,c]
// channel dim contiguous across threads -> coalesced gather + coalesced atomics
template <int NCH>
__global__ void edge_scatter(const int* __restrict__ src, const int* __restrict__ dst,
                             const float* __restrict__ dinv,
                             const float* __restrict__ xw, float* __restrict__ h) {
    int i = blockIdx.x * blockDim.x + threadIdx.x;
    if (i >= N_EDGES * NCH) return;
    int e = i / NCH, c = i % NCH;
    int s = src[e], d = dst[e];
    float norm = dinv[s] * dinv[d];
    atomicAdd(&h[d * NCH + c], norm * xw[s * NCH + c]);
}

template <int NCH>
__global__ void bias_relu(float* __restrict__ h, const float* __restrict__ b) {
    int i = blockIdx.x * blockDim.x + threadIdx.x;
    if (i >= N_NODES * NCH) return;
    int c = i % NCH;
    h[i] = fmaxf(h[i] + b[c], 0.0f);
}

// ---------------- layer-2 GEMM via fp32 WMMA ----------------
// xw2 = h1 @ W2 : (100000 x 64) @ (64 x 128), fp32 end-to-end.
// One wave per 16x16 output tile; 8 waves/block cover the 128 channels of one
// 16-node tile. 16 chained V_WMMA_F32_16X16X4_F32 over K=64.
__global__ __launch_bounds__(256) void gemm2_wmma(const float* __restrict__ h1,
                                                  const float* __restrict__ W2,
                                                  float* __restrict__ xw2) {
    const int lane  = threadIdx.x & 31;
    const int wave  = threadIdx.x >> 5;       // 0..7 -> channel tile
    const int node0 = blockIdx.x * 16;        // node tile base (100000 % 16 == 0)
    const int c0    = wave * 16;              // channel tile base
    const int hi    = lane >> 4;              // 0: lanes 0-15, 1: lanes 16-31
    const int l     = lane & 15;

    v8f acc = {};
#pragma unroll
    for (int k = 0; k < H1C; k += 4) {
        // A 16x4: lane-half selects K pair {k,k+1} vs {k+2,k+3}, row M = l
        const float* arow = h1 + (node0 + l) * H1C + k + hi * 2;
        v2f a; a.x = arow[0]; a.y = arow[1];
        // B 4x16: lane-half selects K pair, col N = l  (W2 row-major [K=64][C=128])
        const float* bcol = W2 + (k + hi * 2) * H2C + c0 + l;
        v2f b; b.x = bcol[0]; b.y = bcol[H2C];
        acc = __builtin_amdgcn_wmma_f32_16x16x4_f32(
            /*neg_a=*/false, a, /*neg_b=*/false, b,
            /*c_mod=*/(short)0, acc, /*reuse_a=*/false, /*reuse_b=*/false);
    }
    // D 16x16 f32: VGPR v -> row M = v + hi*8, col N = l
    float* out = xw2 + (size_t)(node0 + hi * 8) * H2C + c0 + l;
#pragma unroll
    for (int v = 0; v < 8; ++v) out[(size_t)v * H2C] = acc[v];
}

// ---------------- pooling + FC ----------------

__global__ void pool_scatter(const float* __restrict__ h2, const int* __restrict__ batch,
                             float* __restrict__ pool) {
    int i = blockIdx.x * blockDim.x + threadIdx.x;
    if (i >= N_NODES * H2C) return;
    int n = i >> 7, c = i & 127;
    atomicAdd(&pool[batch[n] * H2C + c], h2[i]);
}

__global__ void cnt_scatter(const int* __restrict__ batch, float* __restrict__ cnt) {
    int n = blockIdx.x * blockDim.x + threadIdx.x;
    if (n < N_NODES) atomicAdd(&cnt[batch[n]], 1.0f);
}

__global__ void fc_kernel(const float* __restrict__ pool, const float* __restrict__ cnt,
                          const float* __restrict__ Wfc, const float* __restrict__ bfc,
                          float* __restrict__ out) {
    int g = blockIdx.x * blockDim.x + threadIdx.x;
    if (g >= N_GRAPHS) return;
    float inv = 1.0f / fmaxf(cnt[g], 1.0f);
#pragma unroll
    for (int o = 0; o < OUT_CH; ++o) {
        float acc = bfc[o];
        for (int c = 0; c < H2C; ++c)
            acc += pool[g * H2C + c] * inv * Wfc[c * OUT_CH + o];
        out[g * OUT_CH + o] = acc;
    }
}

// ---------------- launcher ----------------

extern "C" void kernel_launch(void* const* d_in, const int* in_sizes, int n_in,
                              void* d_out, int out_size, void* d_ws, size_t ws_size,
                              hipStream_t stream) {
    const float* x    = (const float*)d_in[0];
    const int*   eidx = (const int*)d_in[1];   // (2, E) flat: [0:E]=src, [E:2E]=dst
    const int*   batch= (const int*)d_in[2];
    const float* W1   = (const float*)d_in[3];
    const float* b1   = (const float*)d_in[4];
    const float* W2   = (const float*)d_in[5];
    const float* b2   = (const float*)d_in[6];
    const float* Wfc  = (const float*)d_in[7];
    const float* bfc  = (const float*)d_in[8];
    float* out = (float*)d_out;

    const int* src = eidx;
    const int* dst = eidx + N_EDGES;

    // workspace layout (floats); bufC is xw1 early, h2 late (xw1 dead before h2 written)
    float* ws   = (float*)d_ws;
    float* dinv = ws;                                          // N (padded)
    float* h1   = dinv + ((N_NODES + 255) & ~255);             // N*64
    float* xw2  = h1 + (size_t)N_NODES * H1C;                  // N*128
    float* bufC = xw2 + (size_t)N_NODES * H2C;                 // N*128 shared
    float* xw1  = bufC;
    float* h2   = bufC;
    float* pool = bufC + (size_t)N_NODES * H2C;                // G*128
    float* cnt  = pool + N_GRAPHS * H2C;                       // G

    auto cdiv = [](long a, long b) { return (int)((a + b - 1) / b); };

    // degree -> dinv (deg starts at 1.0 for the self loop)
    fill_f32<<<cdiv(N_NODES, 256), 256, 0, stream>>>(dinv, 1.0f, N_NODES);
    deg_accum<<<cdiv(N_EDGES, 256), 256, 0, stream>>>(dst, dinv);
    deg_to_dinv<<<cdiv(N_NODES, 256), 256, 0, stream>>>(dinv);

    // layer 1
    xw1_kernel<<<cdiv((long)N_NODES * H1C, 256), 256, 0, stream>>>(x, W1, xw1);
    self_init<H1C><<<cdiv((long)N_NODES * H1C, 256), 256, 0, stream>>>(xw1, dinv, h1);
    edge_scatter<H1C><<<cdiv((long)N_EDGES * H1C, 256), 256, 0, stream>>>(src, dst, dinv, xw1, h1);
    bias_relu<H1C><<<cdiv((long)N_NODES * H1C, 256), 256, 0, stream>>>(h1, b1);

    // layer 2: dense GEMM on WMMA, then aggregation
    gemm2_wmma<<<N_NODES / 16, 256, 0, stream>>>(h1, W2, xw2);
    self_init<H2C><<<cdiv((long)N_NODES * H2C, 256), 256, 0, stream>>>(xw2, dinv, h2);
    edge_scatter<H2C><<<cdiv((long)N_EDGES * H2C, 256), 256, 0, stream>>>(src, dst, dinv, xw2, h2);
    bias_relu<H2C><<<cdiv((long)N_NODES * H2C, 256), 256, 0, stream>>>(h2, b2);

    // mean pool + FC
    fill_f32<<<cdiv(N_GRAPHS * H2C + N_GRAPHS, 256), 256, 0, stream>>>(
        pool, 0.0f, N_GRAPHS * H2C + N_GRAPHS);
    pool_scatter<<<cdiv((long)N_NODES * H2C, 256), 256, 0, stream>>>(h2, batch, pool);
    cnt_scatter<<<cdiv(N_NODES, 256), 256, 0, stream>>>(batch, cnt);
    fc_kernel<<<cdiv(N_GRAPHS, 256), 256, 0, stream>>>(pool, cnt, Wfc, bfc, out);
}